// CMFB_51367808860418
// MI455X (gfx1250) — compile-verified
//
#include <hip/hip_runtime.h>
#include <hip/hip_bf16.h>
#include <math.h>

// ---------------------------------------------------------------------------
// CDNA5 (gfx1250) implementation of the bidirectional CMS6 block.
//   GEMMs  -> v_wmma_f32_16x16x4_f32 (full fp32 precision)
//   Scan   -> double-buffered GLOBAL_LOAD_ASYNC_TO_LDS_B128 staging (ASYNCcnt)
//             + VALU exp/fma recurrence + wave32 shfl reductions
// ---------------------------------------------------------------------------

typedef __attribute__((ext_vector_type(2))) float v2f;
typedef __attribute__((ext_vector_type(4))) float v4f;
typedef __attribute__((ext_vector_type(8))) float v8f;

#define BATCH   8
#define SEQL    4096
#define DMODEL  64
#define DIMC    128
#define DSTATE  64

// ---------------------------------------------------------------------------
// Async copy: 16 bytes global -> LDS per lane, tracked by ASYNCcnt.
// Operand order per CDNA5 VFLAT encoding: VDST = LDS byte address (VGPR),
// then 64-bit global address VGPR pair, saddr = off.
// ---------------------------------------------------------------------------
__device__ __forceinline__ void async_ld_b128(void* lds_ptr, const void* gptr)
{
    unsigned lds_addr = (unsigned)(unsigned long long)lds_ptr;  // low 32 = LDS offset
    asm volatile("global_load_async_to_lds_b128 %0, %1, off"
                 :: "v"(lds_addr), "v"((unsigned long long)gptr)
                 : "memory");
}

template <int N>
__device__ __forceinline__ void wait_asynccnt()
{
    asm volatile("s_wait_asynccnt %0" :: "i"(N) : "memory");
}

// ---------------------------------------------------------------------------
// Generic fp32 WMMA GEMM:  Y[b] = act( W(MxK) * X[b](KxN) + bias )
// storeMode 0: Y[b][m][n]      (row-major, ld = N)
// storeMode 1: Y[b][n][m]      (l-major,   ld = M)  -- scan inputs
// storeMode 2: Y[b][N-1-n][m]  (reversed l)         -- reverse direction
// act 0: none, 1: softplus
// 8 waves / block, each wave owns one 16x16 tile; K consumed 4 at a time.
// ---------------------------------------------------------------------------
__global__ __launch_bounds__(256) void wmma_gemm_f32(
    const float* __restrict__ Wm, const float* __restrict__ bias,
    const float* __restrict__ X, float* __restrict__ Y,
    int M, int K, int N, int strideXb, long long strideYb,
    int act, int storeMode)
{
    const int wave   = threadIdx.x >> 5;
    const int lane   = threadIdx.x & 31;
    const int lane16 = lane & 15;
    const int half   = lane >> 4;          // 0: lanes 0-15, 1: lanes 16-31

    const int mTiles  = M >> 4;
    const int nGroups = (N >> 4) >> 3;     // N tiles / 8 waves
    int bid = blockIdx.x;
    const int b  = bid / (mTiles * nGroups);
    int rem      = bid % (mTiles * nGroups);
    const int mt = rem / nGroups;
    const int ng = rem % nGroups;
    const int nt = (ng << 3) + wave;

    const int m0 = mt << 4;
    const int n0 = nt << 4;
    const int m  = m0 + lane16;            // A row owned by this lane
    const int n  = n0 + lane16;            // B/D column owned by this lane

    // Strength-reduced running pointers (lane's K pair: kk, kk+1)
    const float* wp = Wm + m * K + (half << 1);
    const float* xp = X + (long long)b * strideXb + (long long)(half << 1) * N + n;
    const long long xstep = 4ll * N;

    v8f acc = {};
    for (int k = 0; k < K; k += 4) {
        v2f av = *(const v2f*)wp;          // A: 16x4 tile, row m
        v2f bv;                            // B: 4x16 tile, col n
        bv[0] = xp[0];
        bv[1] = xp[N];
        acc = __builtin_amdgcn_wmma_f32_16x16x4_f32(
                  false, av, false, bv, (short)0, acc, false, false);
        wp += 4;
        xp += xstep;
    }

    // D layout: acc[r] = element (m0 + 8*half + r, n)
    const int mbase = m0 + (half << 3);
    float* Yb = Y + (long long)b * strideYb;

    if (storeMode == 0) {
#pragma unroll
        for (int r = 0; r < 8; ++r) {
            const int mr = mbase + r;
            float v = acc[r];
            if (bias) v += bias[mr];
            if (act == 1) v = (v > 20.0f) ? v : log1pf(__expf(v));
            Yb[(long long)mr * N + n] = v;
        }
    } else {
        const int nst = (storeMode == 1) ? n : (N - 1 - n);
        float* dst = Yb + (long long)nst * M + mbase;   // 8 consecutive floats
        v4f lo, hi;
#pragma unroll
        for (int r = 0; r < 4; ++r) {
            float vl = acc[r],     vh = acc[r + 4];
            if (bias) { vl += bias[mbase + r]; vh += bias[mbase + r + 4]; }
            if (act == 1) {
                vl = (vl > 20.0f) ? vl : log1pf(__expf(vl));
                vh = (vh > 20.0f) ? vh : log1pf(__expf(vh));
            }
            lo[r] = vl; hi[r] = vh;
        }
        *(v4f*)dst       = lo;
        *(v4f*)(dst + 4) = hi;
    }
}

// ---------------------------------------------------------------------------
// W_dt = dt_w (64x4) * yproj_w[0:4] (4x128)  -> 64x128 fused dt projection
// ---------------------------------------------------------------------------
__global__ void build_wdt(const float* __restrict__ dt_w,
                          const float* __restrict__ yproj_w,
                          float* __restrict__ Wdt)
{
    int idx = blockIdx.x * blockDim.x + threadIdx.x;
    if (idx >= DMODEL * DIMC) return;
    int d = idx >> 7, c = idx & 127;
    float s = 0.f;
#pragma unroll
    for (int r = 0; r < 4; ++r)
        s += dt_w[d * 4 + r] * yproj_w[r * DIMC + c];
    Wdt[idx] = s;
}

// ---------------------------------------------------------------------------
// Depthwise causal conv(4) + SiLU, forward and (time-reversed) reverse.
// xcR is stored in reversed-sequence order so the reverse scan runs forward.
// ---------------------------------------------------------------------------
__global__ __launch_bounds__(256) void dwconv_silu(
    const float* __restrict__ xs,
    const float* __restrict__ wf, const float* __restrict__ bf,
    const float* __restrict__ wr, const float* __restrict__ br,
    float* __restrict__ xcF, float* __restrict__ xcR)
{
    const int bd = blockIdx.x;          // b*64 + d
    const int d  = bd & 63;
    const float* row = xs + (long long)bd * SEQL;
    const float w0 = wf[d*4+0], w1 = wf[d*4+1], w2 = wf[d*4+2], w3 = wf[d*4+3];
    const float u0 = wr[d*4+0], u1 = wr[d*4+1], u2 = wr[d*4+2], u3 = wr[d*4+3];
    const float cf = bf[d], cr = br[d];
    for (int l = threadIdx.x; l < SEQL; l += 256) {
        float a = cf;
        if (l >= 3) a += w0 * row[l-3];
        if (l >= 2) a += w1 * row[l-2];
        if (l >= 1) a += w2 * row[l-1];
        a += w3 * row[l];
        xcF[(long long)bd * SEQL + l] = a / (1.f + __expf(-a));   // silu
        // reversed sequence v[j] = row[L-1-j]; causal conv at j = l
        float c = cr;
        if (l >= 3) c += u0 * row[SEQL-1-(l-3)];
        if (l >= 2) c += u1 * row[SEQL-1-(l-2)];
        if (l >= 1) c += u2 * row[SEQL-1-(l-1)];
        c += u3 * row[SEQL-1-l];
        xcR[(long long)bd * SEQL + l] = c / (1.f + __expf(-c));
    }
}

// ---------------------------------------------------------------------------
// Selective-scan. Grid = B * 4 (d-split of 16 channels). 256 threads:
//   thread t: d_local = t/16, n_group = t%16 -> owns 4 states n0..n0+3.
// Double-buffered async LDS staging: every wave issues exactly 5
// global_load_async_to_lds_b128 per chunk (1 dt/u + 2 B + 2 C), so
// s_wait_asynccnt 5 == "previous chunk complete" (async loads retire in order).
// Reduction over n = xor-shuffle across the 16 lanes sharing a channel.
// ---------------------------------------------------------------------------
#define SCH 32
__global__ __launch_bounds__(256) void ssm_scan(
    const float* __restrict__ dt,    // (B, L, 64)
    const float* __restrict__ Bm,    // (B, L, 64)
    const float* __restrict__ Cm,    // (B, L, 64)
    const float* __restrict__ xc,    // (B, 64, L)
    const float* __restrict__ A_log, // (64, 64)
    const float* __restrict__ Dp,    // (64,)
    float* __restrict__ outCat,      // (B, 128, L)
    int dirOff, int reverse)
{
    __shared__ __align__(16) float dt_s[2][SCH * 16];   // [i][dd]
    __shared__ __align__(16) float u_s [2][SCH * 16];   // [dd][i]
    __shared__ __align__(16) float B_s [2][SCH * 64];   // [i][nn]
    __shared__ __align__(16) float C_s [2][SCH * 64];   // [i][nn]
    __shared__ float o_s[SCH * 16];                     // [i][dd]

    const int b  = blockIdx.x >> 2;
    const int ds = blockIdx.x & 3;       // channel split: d in [16*ds, 16*ds+16)
    const int tid = threadIdx.x;
    const int d_local = tid >> 4;
    const int n_group = tid & 15;
    const int n0 = n_group << 2;
    const int d  = (ds << 4) + d_local;

    const float a0 = -__expf(A_log[d * DSTATE + n0 + 0]);
    const float a1 = -__expf(A_log[d * DSTATE + n0 + 1]);
    const float a2 = -__expf(A_log[d * DSTATE + n0 + 2]);
    const float a3 = -__expf(A_log[d * DSTATE + n0 + 3]);
    const float Dv = Dp[d];
    float h0 = 0.f, h1 = 0.f, h2 = 0.f, h3 = 0.f;

    const float* dtB = dt + (long long)b * SEQL * 64;
    const float* BB  = Bm + (long long)b * SEQL * 64;
    const float* CB  = Cm + (long long)b * SEQL * 64;
    const float* xcB = xc + ((long long)b * 64 + (ds << 4)) * SEQL;
    float* outB = outCat + ((long long)b * DIMC + dirOff + (ds << 4)) * SEQL;

    // Issue one chunk's async loads: uniform 5 b128 ops per thread/wave.
    auto issue = [&](int c, int bufi) {
        const int l0 = c * SCH;
        if (tid < 128) {                         // dt: 32x16 floats, 128 ops
            int i  = tid >> 2;
            int dd = (tid & 3) << 2;
            async_ld_b128(&dt_s[bufi][i * 16 + dd],
                          dtB + (long long)(l0 + i) * 64 + (ds << 4) + dd);
        } else {                                 // u: 16 rows x 32, 128 ops
            int t  = tid - 128;
            int dd = t >> 3;
            int i4 = (t & 7) << 2;
            async_ld_b128(&u_s[bufi][dd * SCH + i4],
                          xcB + (long long)dd * SEQL + l0 + i4);
        }
#pragma unroll
        for (int k2 = 0; k2 < 2; ++k2) {         // B: 2048 floats, 512 ops
            int fl = (tid + (k2 << 8)) << 2;
            int i = fl >> 6, nn = fl & 63;
            async_ld_b128(&B_s[bufi][fl], BB + (long long)(l0 + i) * 64 + nn);
        }
#pragma unroll
        for (int k2 = 0; k2 < 2; ++k2) {         // C: 2048 floats, 512 ops
            int fl = (tid + (k2 << 8)) << 2;
            int i = fl >> 6, nn = fl & 63;
            async_ld_b128(&C_s[bufi][fl], CB + (long long)(l0 + i) * 64 + nn);
        }
    };

    const int NCH = SEQL / SCH;                  // 128 chunks
    issue(0, 0);
    for (int c = 0; c < NCH; ++c) {
        const int cur = c & 1;
        if (c + 1 < NCH) {
            issue(c + 1, cur ^ 1);               // prefetch next chunk
            wait_asynccnt<5>();                  // this chunk's 5 ops done
        } else {
            wait_asynccnt<0>();
        }
        __syncthreads();                         // all waves' loads visible

        for (int i = 0; i < SCH; ++i) {
            const float dtv = dt_s[cur][i * 16 + d_local];
            const float uv  = u_s [cur][d_local * SCH + i];
            const float du  = dtv * uv;
            const float* Bi = &B_s[cur][i * 64 + n0];
            const float* Ci = &C_s[cur][i * 64 + n0];
            h0 = __expf(dtv * a0) * h0 + du * Bi[0];
            h1 = __expf(dtv * a1) * h1 + du * Bi[1];
            h2 = __expf(dtv * a2) * h2 + du * Bi[2];
            h3 = __expf(dtv * a3) * h3 + du * Bi[3];
            float po = h0 * Ci[0] + h1 * Ci[1] + h2 * Ci[2] + h3 * Ci[3];
            po += __shfl_xor(po, 1, 16);
            po += __shfl_xor(po, 2, 16);
            po += __shfl_xor(po, 4, 16);
            po += __shfl_xor(po, 8, 16);
            if (n_group == 0) o_s[i * 16 + d_local] = po + Dv * uv;
        }
        __syncthreads();                         // o_s ready; buf reusable

        for (int idx = tid; idx < SCH * 16; idx += 256) {
            int dd = idx >> 5, i = idx & 31;
            int l  = c * SCH + i;
            int lo = reverse ? (SEQL - 1 - l) : l;
            outB[(long long)dd * SEQL + lo] = o_s[i * 16 + dd];
        }
        __syncthreads();
    }
}

// ---------------------------------------------------------------------------
extern "C" void kernel_launch(void* const* d_in, const int* in_sizes, int n_in,
                              void* d_out, int out_size, void* d_ws, size_t ws_size,
                              hipStream_t stream)
{
    const float* x        = (const float*)d_in[0];
    const float* y        = (const float*)d_in[1];
    const float* to_x_w   = (const float*)d_in[2];
    const float* to_x_b   = (const float*)d_in[3];
    const float* to_y_w   = (const float*)d_in[4];
    const float* to_y_b   = (const float*)d_in[5];
    const float* proj_w   = (const float*)d_in[6];
    const float* proj_b   = (const float*)d_in[7];
    const float* f_conv_w = (const float*)d_in[8];
    const float* f_conv_b = (const float*)d_in[9];
    const float* f_yproj  = (const float*)d_in[10];
    const float* f_dt_w   = (const float*)d_in[11];
    const float* f_dt_b   = (const float*)d_in[12];
    const float* f_A_log  = (const float*)d_in[13];
    const float* f_D      = (const float*)d_in[14];
    const float* r_conv_w = (const float*)d_in[15];
    const float* r_conv_b = (const float*)d_in[16];
    const float* r_yproj  = (const float*)d_in[17];
    const float* r_dt_w   = (const float*)d_in[18];
    const float* r_dt_b   = (const float*)d_in[19];
    const float* r_A_log  = (const float*)d_in[20];
    const float* r_D      = (const float*)d_in[21];
    float* out = (float*)d_out;

    // Workspace carve-up (floats)
    float* ws = (float*)d_ws;
    size_t off = 0;
    const size_t szBdL  = (size_t)BATCH * DMODEL * SEQL;  // (B,64,L)
    const size_t szBcL  = (size_t)BATCH * DIMC * SEQL;    // (B,128,L)
    float* xs   = ws + off; off += szBdL;
    float* ys   = ws + off; off += szBcL;
    float* xcF  = ws + off; off += szBdL;
    float* xcR  = ws + off; off += szBdL;
    float* dtF  = ws + off; off += szBdL;   // (B,L,64)
    float* dtR  = ws + off; off += szBdL;
    float* Bf   = ws + off; off += szBdL;
    float* Br   = ws + off; off += szBdL;
    float* Cf   = ws + off; off += szBdL;
    float* Cr   = ws + off; off += szBdL;
    float* cat  = ws + off; off += szBcL;
    float* WdtF = ws + off; off += DMODEL * DIMC;
    float* WdtR = ws + off; off += DMODEL * DIMC;

    const int N = SEQL;
    const int strideX128 = DIMC * N;      // (128, L) per batch
    const long long sY64  = (long long)DMODEL * N;
    const long long sY128 = (long long)DIMC * N;

    // Fused dt-projection weights
    build_wdt<<<32, 256, 0, stream>>>(f_dt_w, f_yproj, WdtF);
    build_wdt<<<32, 256, 0, stream>>>(r_dt_w, r_yproj, WdtR);

    // xs = to_x_w * x ; ys = to_y_w * y
    {
        int blk = BATCH * (DMODEL / 16) * ((N / 16) / 8);   // 1024
        wmma_gemm_f32<<<blk, 256, 0, stream>>>(to_x_w, to_x_b, x, xs,
            DMODEL, DIMC, N, strideX128, sY64, 0, 0);
    }
    {
        int blk = BATCH * (DIMC / 16) * ((N / 16) / 8);     // 2048
        wmma_gemm_f32<<<blk, 256, 0, stream>>>(to_y_w, to_y_b, y, ys,
            DIMC, DIMC, N, strideX128, sY128, 0, 0);
    }

    // Depthwise causal convs + SiLU (forward and reversed-order)
    dwconv_silu<<<BATCH * DMODEL, 256, 0, stream>>>(
        xs, f_conv_w, f_conv_b, r_conv_w, r_conv_b, xcF, xcR);

    // B/C/dt projections from ys (reverse direction stored time-reversed)
    {
        int blk = BATCH * (DMODEL / 16) * ((N / 16) / 8);   // 1024
        wmma_gemm_f32<<<blk, 256, 0, stream>>>(f_yproj + 4 * DIMC,  nullptr, ys, Bf,
            DMODEL, DIMC, N, strideX128, sY64, 0, 1);
        wmma_gemm_f32<<<blk, 256, 0, stream>>>(f_yproj + 68 * DIMC, nullptr, ys, Cf,
            DMODEL, DIMC, N, strideX128, sY64, 0, 1);
        wmma_gemm_f32<<<blk, 256, 0, stream>>>(r_yproj + 4 * DIMC,  nullptr, ys, Br,
            DMODEL, DIMC, N, strideX128, sY64, 0, 2);
        wmma_gemm_f32<<<blk, 256, 0, stream>>>(r_yproj + 68 * DIMC, nullptr, ys, Cr,
            DMODEL, DIMC, N, strideX128, sY64, 0, 2);
        wmma_gemm_f32<<<blk, 256, 0, stream>>>(WdtF, f_dt_b, ys, dtF,
            DMODEL, DIMC, N, strideX128, sY64, 1, 1);
        wmma_gemm_f32<<<blk, 256, 0, stream>>>(WdtR, r_dt_b, ys, dtR,
            DMODEL, DIMC, N, strideX128, sY64, 1, 2);
    }

    // Sequential selective scans (forward into cat[:,0:64], reverse into [:,64:128])
    ssm_scan<<<BATCH * 4, 256, 0, stream>>>(dtF, Bf, Cf, xcF, f_A_log, f_D, cat, 0, 0);
    ssm_scan<<<BATCH * 4, 256, 0, stream>>>(dtR, Br, Cr, xcR, r_A_log, r_D, cat, DMODEL, 1);

    // Final projection: out = proj_w * cat + proj_b
    {
        int blk = BATCH * (DIMC / 16) * ((N / 16) / 8);     // 2048
        wmma_gemm_f32<<<blk, 256, 0, stream>>>(proj_w, proj_b, cat, out,
            DIMC, DIMC, N, strideX128, sY128, 0, 0);
    }
}